// GraphVertExtraLinModel_23914377904736
// MI455X (gfx1250) — compile-verified
//
#include <hip/hip_runtime.h>
#include <hip/hip_bf16.h>

typedef __attribute__((ext_vector_type(2))) float v2f;
typedef __attribute__((ext_vector_type(8))) float v8f;

#define NGRAPH 128
#define GS 4
#define MM 64
#define DD 512
#define LIND 128

// ---------------------------------------------------------------------------
// One GML layer:  h_out[n,m,p] = max_g relu( G[n,g] @ (h_in[n] @ W[g]^T + b[g]) )  (+ h_in if RES)
// Grid: (N, D/PT). Block: 128 threads = 4 waves, wave w handles group g = w.
// PT = 64 output columns per block: 4x4 accumulator tiles per wave,
// 8 fragment loads per 16 WMMAs in the K loop.
// ---------------------------------------------------------------------------
template<int C, bool RES>
__global__ __launch_bounds__(128) void gml_layer_kernel(
    const float* __restrict__ G,    // [N, GS, 64, 64]
    const float* __restrict__ hin,  // [N, 64, C]
    const float* __restrict__ W,    // [GS, 512, C]
    const float* __restrict__ bias, // [GS, 512]
    float* __restrict__ hout)       // [N, 64, 512]
{
  constexpr int PT = 64;
  __shared__ float lds[GS][MM][PT];   // 4*64*64*4 = 65,536 B (exactly 64 KB)

  const int n     = blockIdx.x;
  const int pBase = blockIdx.y * PT;
  const int g     = threadIdx.x >> 5;     // wave id == group id
  const int lane  = threadIdx.x & 31;
  const int l16   = lane & 15;
  const int coff  = (lane >> 4) * 2;      // lanes 0-15 -> K+0/K+1, lanes 16-31 -> K+2/K+3
  const int hi8   = (lane < 16) ? 0 : 8;  // C/D row offset for upper lane half

  const float* hN = hin + (size_t)n * MM * C;
  const float* Wg = W + (size_t)g * DD * C;

  // ---------------- Stage 1: multi = h @ W^T  (64 x PT tile) ----------------
  v8f acc[4][4];
  #pragma unroll
  for (int mt = 0; mt < 4; ++mt)
    #pragma unroll
    for (int pt = 0; pt < 4; ++pt)
      acc[mt][pt] = {};

  for (int c = 0; c < C; c += 4) {
    v2f a[4];
    #pragma unroll
    for (int mt = 0; mt < 4; ++mt)
      a[mt] = *(const v2f*)(hN + (size_t)(mt * 16 + l16) * C + c + coff);
    #pragma unroll
    for (int pt = 0; pt < 4; ++pt) {
      // B[k=c][col=p] = W[g, p, c]  -> one float2 per lane
      v2f bf = *(const v2f*)(Wg + (size_t)(pBase + pt * 16 + l16) * C + c + coff);
      #pragma unroll
      for (int mt = 0; mt < 4; ++mt)
        acc[mt][pt] = __builtin_amdgcn_wmma_f32_16x16x4_f32(
            false, a[mt], false, bf, (short)0, acc[mt][pt], false, false);
    }
  }

  // bias add + spill multi to LDS (C/D layout -> [row][col])
  #pragma unroll
  for (int pt = 0; pt < 4; ++pt) {
    float bb = bias[g * DD + pBase + pt * 16 + l16];
    #pragma unroll
    for (int mt = 0; mt < 4; ++mt) {
      #pragma unroll
      for (int v = 0; v < 8; ++v) {
        int row = mt * 16 + v + hi8;
        lds[g][row][pt * 16 + l16] = acc[mt][pt][v] + bb;
      }
    }
  }
  __syncthreads();

  // ---------------- Stage 2: xo = G[n,g] @ multi ----------------
  v8f acc2[4][4];
  #pragma unroll
  for (int mt = 0; mt < 4; ++mt)
    #pragma unroll
    for (int pt = 0; pt < 4; ++pt)
      acc2[mt][pt] = {};

  const float* Gn = G + ((size_t)n * GS + g) * MM * MM;
  for (int k = 0; k < MM; k += 4) {
    v2f a[4];
    #pragma unroll
    for (int mt = 0; mt < 4; ++mt)
      a[mt] = *(const v2f*)(Gn + (size_t)(mt * 16 + l16) * MM + k + coff);
    #pragma unroll
    for (int pt = 0; pt < 4; ++pt) {
      v2f bf;
      bf.x = lds[g][k + coff][pt * 16 + l16];
      bf.y = lds[g][k + coff + 1][pt * 16 + l16];
      #pragma unroll
      for (int mt = 0; mt < 4; ++mt)
        acc2[mt][pt] = __builtin_amdgcn_wmma_f32_16x16x4_f32(
            false, a[mt], false, bf, (short)0, acc2[mt][pt], false, false);
    }
  }
  __syncthreads();

  // xo -> LDS (per-g region)
  #pragma unroll
  for (int pt = 0; pt < 4; ++pt)
    #pragma unroll
    for (int mt = 0; mt < 4; ++mt)
      #pragma unroll
      for (int v = 0; v < 8; ++v) {
        int row = mt * 16 + v + hi8;
        lds[g][row][pt * 16 + l16] = acc2[mt][pt][v];
      }
  __syncthreads();

  // relu(max_g) + residual + store (relu and max commute)
  for (int e = threadIdx.x; e < MM * PT; e += 128) {
    int r = e >> 6;         // PT = 64
    int c = e & 63;
    float v = fmaxf(fmaxf(lds[0][r][c], lds[1][r][c]),
                    fmaxf(lds[2][r][c], lds[3][r][c]));
    v = fmaxf(v, 0.0f);
    size_t oi = ((size_t)n * MM + r) * DD + pBase + c;
    if (RES) v += hin[oi];   // only used when C == D
    hout[oi] = v;
  }
}

// ---------------------------------------------------------------------------
// Head: x1 = relu(h @ lin1_w^T + b1)  [64x128];  out = x1 @ lin2_w^T + b2  [64x1]
// Grid: (N). Block: 128 threads = 4 waves; wave w covers output cols 32w..32w+31.
// ---------------------------------------------------------------------------
__global__ __launch_bounds__(128) void head_kernel(
    const float* __restrict__ h,   // [N, 64, 512]
    const float* __restrict__ w1,  // [128, 512]
    const float* __restrict__ b1,  // [128]
    const float* __restrict__ w2,  // [1, 128]
    const float* __restrict__ b2,  // [1]
    float* __restrict__ out)       // [N, 64]
{
  __shared__ float lds[MM][LIND + 1];   // 64*129*4 = 33,024 B

  const int n     = blockIdx.x;
  const int wave  = threadIdx.x >> 5;
  const int lane  = threadIdx.x & 31;
  const int l16   = lane & 15;
  const int coff  = (lane >> 4) * 2;
  const int hi8   = (lane < 16) ? 0 : 8;
  const int eBase = wave * 32;

  const float* hN = h + (size_t)n * MM * DD;

  v8f acc[4][2];
  #pragma unroll
  for (int mt = 0; mt < 4; ++mt)
    #pragma unroll
    for (int pt = 0; pt < 2; ++pt)
      acc[mt][pt] = {};

  for (int c = 0; c < DD; c += 4) {
    v2f a[4];
    #pragma unroll
    for (int mt = 0; mt < 4; ++mt)
      a[mt] = *(const v2f*)(hN + (size_t)(mt * 16 + l16) * DD + c + coff);
    #pragma unroll
    for (int pt = 0; pt < 2; ++pt) {
      v2f bf = *(const v2f*)(w1 + (size_t)(eBase + pt * 16 + l16) * DD + c + coff);
      #pragma unroll
      for (int mt = 0; mt < 4; ++mt)
        acc[mt][pt] = __builtin_amdgcn_wmma_f32_16x16x4_f32(
            false, a[mt], false, bf, (short)0, acc[mt][pt], false, false);
    }
  }

  #pragma unroll
  for (int pt = 0; pt < 2; ++pt) {
    float bb = b1[eBase + pt * 16 + l16];
    #pragma unroll
    for (int mt = 0; mt < 4; ++mt)
      #pragma unroll
      for (int v = 0; v < 8; ++v) {
        int row = mt * 16 + v + hi8;
        lds[row][eBase + pt * 16 + l16] = fmaxf(acc[mt][pt][v] + bb, 0.0f);
      }
  }
  __syncthreads();

  for (int m = threadIdx.x; m < MM; m += 128) {
    float s = 0.0f;
    #pragma unroll 8
    for (int e = 0; e < LIND; ++e) s += lds[m][e] * w2[e];
    out[(size_t)n * MM + m] = s + b2[0];
  }
}

// ---------------------------------------------------------------------------
extern "C" void kernel_launch(void* const* d_in, const int* in_sizes, int n_in,
                              void* d_out, int out_size, void* d_ws, size_t ws_size,
                              hipStream_t stream) {
  const float* G   = (const float*)d_in[0];  // [128,4,64,64]
  const float* x   = (const float*)d_in[1];  // [128,64,128]
  const float* W0  = (const float*)d_in[2];  // [4,512,128]
  const float* b0  = (const float*)d_in[3];  // [4,512]
  const float* W   = (const float*)d_in[4];  // [7,4,512,512]
  const float* b   = (const float*)d_in[5];  // [7,4,512]
  const float* w1  = (const float*)d_in[6];  // [128,512]
  const float* b1  = (const float*)d_in[7];  // [128]
  const float* w2  = (const float*)d_in[8];  // [1,128]
  const float* b2  = (const float*)d_in[9];  // [1]
  float* out = (float*)d_out;

  float* hA = (float*)d_ws;                         // [128,64,512]
  float* hB = hA + (size_t)NGRAPH * MM * DD;        // [128,64,512]

  dim3 blk(128);
  dim3 grid(NGRAPH, DD / 64);

  gml_layer_kernel<128, false><<<grid, blk, 0, stream>>>(G, x, W0, b0, hA);

  float* cur = hA;
  float* nxt = hB;
  for (int l = 0; l < 7; ++l) {
    gml_layer_kernel<512, true><<<grid, blk, 0, stream>>>(
        G, cur, W + (size_t)l * GS * DD * DD, b + (size_t)l * GS * DD, nxt);
    float* t = cur; cur = nxt; nxt = t;
  }

  head_kernel<<<dim3(NGRAPH), blk, 0, stream>>>(cur, w1, b1, w2, b2, out);
}